// NCCLoss_38019050504637
// MI455X (gfx1250) — compile-verified
//
#include <hip/hip_runtime.h>

// NCC loss (local normalized cross-correlation, 9x9x9 window, SAME zero-pad)
// over (2,1,128,128,128) fp32 volumes, for MI455X / gfx1250.
//
// Pass 1: pointwise products + x-axis box sum        (HBM read 32MB, write 80MB)
// Pass 2: y-axis box sum (running window)            (L2-resident 80MB -> 80MB)
// Pass 3: z-axis box sum as banded GEMM on fp32 WMMA (V_WMMA_F32_16X16X4_F32)
//         fused with cc-map pointwise + reduction.
//         Boundary handling folded into the in-register band matrix A so all
//         global loads are unconditional (no exec-mask branches, full MLP).
// Pass 4: deterministic fixed-order final reduction -> scalar loss

typedef __attribute__((ext_vector_type(2))) float v2f;
typedef __attribute__((ext_vector_type(8))) float v8f;

#define NX   128
#define ZSTR (128 * 128)                     // z stride in elements
#define VOL  ((size_t)2 * 128 * 128 * 128)   // elements per channel = 4,194,304
#define INV_WIN (1.0f / 729.0f)
#define NCC_EPS 1e-5f

// ---------------- Kernel 1 : products + X-axis 9-tap box sum ----------------
__global__ __launch_bounds__(128)
void ncc_k1_xpass(const float* __restrict__ J,   // y_pred
                  const float* __restrict__ I,   // y_true
                  float* __restrict__ out)       // 5 channels, VOL each
{
    __shared__ float sI[NX];
    __shared__ float sJ[NX];
    const int    x    = threadIdx.x;
    const size_t base = (size_t)blockIdx.x * NX;   // one (b,z,y) row per block
    sI[x] = I[base + x];
    sJ[x] = J[base + x];
    __syncthreads();

    float s0 = 0.f, s1 = 0.f, s2 = 0.f, s3 = 0.f, s4 = 0.f;
#pragma unroll
    for (int t = -4; t <= 4; ++t) {
        const int k = x + t;
        if (k >= 0 && k < NX) {
            const float i = sI[k], j = sJ[k];
            s0 += i; s1 += j; s2 += i * i; s3 += j * j; s4 += i * j;
        }
    }
    const size_t idx = base + x;
    out[0 * VOL + idx] = s0;   // sum I
    out[1 * VOL + idx] = s1;   // sum J
    out[2 * VOL + idx] = s2;   // sum I^2
    out[3 * VOL + idx] = s3;   // sum J^2
    out[4 * VOL + idx] = s4;   // sum I*J
}

// ---------------- Kernel 2 : Y-axis 9-tap box sum (running window) ----------
__global__ __launch_bounds__(128)
void ncc_k2_ypass(const float* __restrict__ in, float* __restrict__ out)
{
    const int x = threadIdx.x;
    int id = blockIdx.x;                 // ((c*2 + b)*128 + z)
    const int z = id & 127; id >>= 7;
    const int b = id & 1;   id >>= 1;
    const int c = id;                    // 0..4
    const size_t base = (size_t)c * VOL + (size_t)(b * 128 + z) * ZSTR + x;

    float s = 0.f;
#pragma unroll
    for (int k = 0; k < 4; ++k) s += in[base + (size_t)k * NX];
    for (int y = 0; y < 128; ++y) {
        if (y + 4 < 128) s += in[base + (size_t)(y + 4) * NX];
        out[base + (size_t)y * NX] = s;
        if (y - 4 >= 0) s -= in[base + (size_t)(y - 4) * NX];
    }
}

// ------- Kernel 3 : Z-axis box sum as banded GEMM via fp32 WMMA + cc-map ----
// One wave per 16(z) x 16(x) output tile at fixed (b, y).
// C[m,n] = sum_k Band[z0+m, k] * in[k, x0+n],  Band[z,k] = (|z-k| <= 4).
// A (16x4 band chunk) is synthesized in registers. Zero-pad boundary is folded
// into A: rows k outside [0,128) get A==0, and the corresponding B load
// address is clamped in-range, so every global load is unconditional.
__global__ __launch_bounds__(256)
void ncc_k3_zpass_wmma(const float* __restrict__ ws2,   // 5ch, x/y-filtered
                       float* __restrict__ partial)     // one partial per block
{
    __shared__ float red[256];
    const int tid  = threadIdx.x;
    const int lane = tid & 31;
    const int wave = tid >> 5;
    const int half = lane >> 4;     // 0: lanes 0-15, 1: lanes 16-31
    const int mn   = lane & 15;     // A row (M) == B/C column (N) index

    int tile = blockIdx.x * 8 + wave;       // 16384 tiles total
    const int xb = tile & 7;   tile >>= 3;
    const int zb = tile & 7;   tile >>= 3;
    const int y  = tile & 127; tile >>= 7;
    const int b  = tile;                    // 0..1
    const int z0 = zb * 16;
    const int x0 = xb * 16;

    // element address at (b, z=0, y, x0+n)
    const size_t off = (size_t)b * (128 * ZSTR) + (size_t)y * NX + (x0 + mn);

    v8f acc0 = {}, acc1 = {}, acc2 = {}, acc3 = {}, acc4 = {};

#pragma unroll
    for (int kc = 0; kc < 6; ++kc) {
        const int k0 = z0 - 4 + kc * 4;     // K-chunk base, covers z0-4 .. z0+19
        const int kb = k0 + half * 2;       // first global-z this lane holds

        // Band-matrix chunk A (16x4): VGPR0 holds K={0,2}, VGPR1 K={1,3} split
        // across lane halves; lanes 0-15 carry M=0..15. Validity of the global
        // z-row is ANDed into the band predicate (realizes the zero pad).
        const int d0 = kb - (z0 + mn);
        const int d1 = d0 + 1;
        const bool in0 = (kb >= 0) & (kb < 128);
        const bool in1 = (kb + 1 >= 0) & (kb + 1 < 128);
        v2f a;
        a.x = (d0 >= -4 && d0 <= 4 && in0) ? 1.f : 0.f;
        a.y = (d1 >= -4 && d1 <= 4 && in1) ? 1.f : 0.f;

        // Clamped (always-valid) addresses for the two z-rows this lane holds.
        const int r0 = min(max(kb, 0), 127);
        const int r1 = min(max(kb + 1, 0), 127);
        const size_t o0 = off + (size_t)r0 * ZSTR;
        const size_t o1 = off + (size_t)r1 * ZSTR;

        v2f bb;
        bb.x = ws2[0 * VOL + o0];
        bb.y = ws2[0 * VOL + o1];
        acc0 = __builtin_amdgcn_wmma_f32_16x16x4_f32(false, a, false, bb,
                                                     (short)0, acc0, false, false);
        bb.x = ws2[1 * VOL + o0];
        bb.y = ws2[1 * VOL + o1];
        acc1 = __builtin_amdgcn_wmma_f32_16x16x4_f32(false, a, false, bb,
                                                     (short)0, acc1, false, false);
        bb.x = ws2[2 * VOL + o0];
        bb.y = ws2[2 * VOL + o1];
        acc2 = __builtin_amdgcn_wmma_f32_16x16x4_f32(false, a, false, bb,
                                                     (short)0, acc2, false, false);
        bb.x = ws2[3 * VOL + o0];
        bb.y = ws2[3 * VOL + o1];
        acc3 = __builtin_amdgcn_wmma_f32_16x16x4_f32(false, a, false, bb,
                                                     (short)0, acc3, false, false);
        bb.x = ws2[4 * VOL + o0];
        bb.y = ws2[4 * VOL + o1];
        acc4 = __builtin_amdgcn_wmma_f32_16x16x4_f32(false, a, false, bb,
                                                     (short)0, acc4, false, false);
    }

    // Pointwise cc-map over this lane's 8 C entries (C layout: VGPR r holds
    // M=r (lanes 0-15) / M=r+8 (lanes 16-31), N = lane).
    float local = 0.f;
#pragma unroll
    for (int r = 0; r < 8; ++r) {
        const float Is = acc0[r], Js = acc1[r];
        const float I2 = acc2[r], J2 = acc3[r], IJ = acc4[r];
        const float cross = IJ - Is * Js * INV_WIN;
        const float Ivar  = I2 - Is * Is * INV_WIN;
        const float Jvar  = J2 - Js * Js * INV_WIN;
        local += (cross * cross) / (Ivar * Jvar + NCC_EPS);
    }

    // Deterministic fixed-order block reduction.
    red[tid] = local;
    __syncthreads();
    if (tid == 0) {
        float t = 0.f;
        for (int i = 0; i < 256; ++i) t += red[i];
        partial[blockIdx.x] = t;
    }
}

// ---------------- Kernel 4 : final deterministic reduction ------------------
__global__ void ncc_k4_reduce(const float* __restrict__ partial,
                              float* __restrict__ out)
{
    float t = 0.f;
    for (int i = 0; i < 2048; ++i) t += partial[i];
    out[0] = -t / (float)VOL;    // mean(-cc_map)
}

extern "C" void kernel_launch(void* const* d_in, const int* in_sizes, int n_in,
                              void* d_out, int out_size, void* d_ws, size_t ws_size,
                              hipStream_t stream) {
    (void)in_sizes; (void)n_in; (void)out_size; (void)ws_size;
    const float* J = (const float*)d_in[0];   // y_pred
    const float* I = (const float*)d_in[1];   // y_true
    float* out = (float*)d_out;

    float* ws1     = (float*)d_ws;            // 5 * VOL floats (80 MB)
    float* ws2     = ws1 + 5 * VOL;           // 5 * VOL floats (80 MB)
    float* partial = ws2 + 5 * VOL;           // 2048 floats

    ncc_k1_xpass<<<2 * 128 * 128, 128, 0, stream>>>(J, I, ws1);
    ncc_k2_ypass<<<5 * 2 * 128, 128, 0, stream>>>(ws1, ws2);
    ncc_k3_zpass_wmma<<<2048, 256, 0, stream>>>(ws2, partial);
    ncc_k4_reduce<<<1, 1, 0, stream>>>(partial, out);
}